// DecoderAttention_8890582302735
// MI455X (gfx1250) — compile-verified
//
#include <hip/hip_runtime.h>
#include <hip/hip_bf16.h>
#include <math.h>

// Problem dims (match reference)
#define BB   32
#define LL   128
#define HH   1024
#define ENN  256
#define ETT  512
#define EIN  768
#define G4H  4096           // 4*H
#define VOUT 50003
#define K3H  3072           // 3*H
#define ROWOUT (VOUT + LL)  // 50131 logits row length

typedef __attribute__((ext_vector_type(2))) float v2f;
typedef __attribute__((ext_vector_type(8))) float v8f;

__device__ __forceinline__ v2f ld2(const float* p) { return *(const v2f*)p; }

// Non-temporal load for one-shot weight streams (TH=NT): don't pollute the
// caches that hold reused activations (s3, enc_out, z).
__device__ __forceinline__ v2f ld2_nt(const float* p) {
  return __builtin_nontemporal_load((const v2f*)p);
}

// D(16x16,f32) += A(16x4,f32) * B(4x16,f32), CDNA5 wave32 WMMA
__device__ __forceinline__ v8f wmma4(v2f a, v2f b, v8f c) {
  return __builtin_amdgcn_wmma_f32_16x16x4_f32(false, a, false, b, (short)0, c,
                                               false, false);
}

// Inner loop: two M-tiles (rows 0..15 and 16..31) sharing one weight fragment.
// A: [32, K] row-major (cached loads), W: [N, K] row-major (NT streaming).
// Computes A @ W^T for columns n0..n0+15 (lane r supplies column n0+r).
__device__ __forceinline__ void gemm2_acc(const float* A, int lda,
                                          const float* Wrow, int K,
                                          int half, int r,
                                          v8f& acc0, v8f& acc1) {
  for (int k = 0; k < K; k += 4) {
    v2f bf = ld2_nt(Wrow + k + 2 * half);              // weight stream, NT
    v2f a0 = ld2(A + (size_t)r * lda + k + 2 * half);        // rows 0..15
    v2f a1 = ld2(A + (size_t)(16 + r) * lda + k + 2 * half); // rows 16..31
    acc0 = wmma4(a0, bf, acc0);
    acc1 = wmma4(a1, bf, acc1);
  }
}

// ---------------- Kernel A: embedding gather + concat -> x [32,768]
__global__ void k_embed(const int* n_in, const int* t_in,
                        const float* embN, const float* embT, float* x) {
  int idx = blockIdx.x * blockDim.x + threadIdx.x;
  if (idx >= BB * EIN) return;
  int b = idx / EIN, j = idx % EIN;
  float v = (j < ENN) ? embN[(size_t)n_in[b] * ENN + j]
                      : embT[(size_t)t_in[b] * ETT + (j - ENN)];
  x[idx] = v;
}

// ---------------- Kernel B: gates = x@Wih^T + h0@Whh^T + bih + bhh  [32,4096]
__global__ void k_gates(const float* x, const float* hprev,
                        const float* Wih, const float* Whh,
                        const float* bih, const float* bhh, float* gates) {
  int wave = (blockIdx.x * blockDim.x + threadIdx.x) >> 5;
  int lane = threadIdx.x & 31;
  if (wave >= G4H / 16) return;
  int half = lane >> 4, r = lane & 15;
  int n = wave * 16 + r;
  float bias = bih[n] + bhh[n];
  v8f acc0, acc1;
  for (int i = 0; i < 8; i++) { acc0[i] = bias; acc1[i] = bias; }
  gemm2_acc(x, EIN, Wih + (size_t)n * EIN, EIN, half, r, acc0, acc1);
  gemm2_acc(hprev, HH, Whh + (size_t)n * HH, HH, half, r, acc0, acc1);
  for (int v = 0; v < 8; v++) {
    int m = v + 8 * half;
    gates[(size_t)m * G4H + n] = acc0[v];
    gates[(size_t)(16 + m) * G4H + n] = acc1[v];
  }
}

// ---------------- Kernel C: LSTM cell (i,f,g,o PyTorch order)
__global__ void k_cell(const float* gates, const float* c0,
                       float* hnew, float* hout, float* cout) {
  int idx = blockIdx.x * blockDim.x + threadIdx.x;
  if (idx >= BB * HH) return;
  int b = idx / HH, h = idx % HH;
  const float* gr = gates + (size_t)b * G4H;
  float ig = gr[h], fg = gr[HH + h], gg = gr[2 * HH + h], og = gr[3 * HH + h];
  float si = 1.f / (1.f + expf(-ig));
  float sf = 1.f / (1.f + expf(-fg));
  float so = 1.f / (1.f + expf(-og));
  float c = sf * c0[idx] + si * tanhf(gg);
  float hh = so * tanhf(c);
  hnew[idx] = hh; hout[idx] = hh; cout[idx] = c;
}

// ---------------- Kernel D: q = hnew@Whw^T + Whb  [32,1024]
__global__ void k_query(const float* hnew, const float* Whw, const float* Whb,
                        float* q) {
  int wave = (blockIdx.x * blockDim.x + threadIdx.x) >> 5;
  int lane = threadIdx.x & 31;
  if (wave >= HH / 16) return;
  int half = lane >> 4, r = lane & 15;
  int n = wave * 16 + r;
  float bias = Whb[n];
  v8f acc0, acc1;
  for (int i = 0; i < 8; i++) { acc0[i] = bias; acc1[i] = bias; }
  gemm2_acc(hnew, HH, Whw + (size_t)n * HH, HH, half, r, acc0, acc1);
  for (int v = 0; v < 8; v++) {
    int m = v + 8 * half;
    q[(size_t)m * HH + n] = acc0[v];
    q[(size_t)(16 + m) * HH + n] = acc1[v];
  }
}

// ---------------- Kernel E: keys = enc@Wmw^T + Wmb  [4096,1024] (big WMMA GEMM)
__global__ void k_keys(const float* enc, const float* Wmw, const float* Wmb,
                       float* keys) {
  int wave = (blockIdx.x * blockDim.x + threadIdx.x) >> 5;
  int lane = threadIdx.x & 31;
  int mt = wave >> 6, nt = wave & 63;   // 256 x 64 tiles
  if (mt >= (BB * LL) / 16) return;
  int half = lane >> 4, r = lane & 15;
  int n = nt * 16 + r;
  v8f acc;
  float bias = Wmb[n];
  for (int i = 0; i < 8; i++) acc[i] = bias;
  const float* wrow = Wmw + (size_t)n * HH;
  const float* arow = enc + (size_t)(mt * 16 + r) * HH;
  for (int k = 0; k < HH; k += 4) {
    v2f bf = ld2_nt(wrow + k + 2 * half);   // weight stream, NT
    v2f a  = ld2(arow + k + 2 * half);      // enc_out reused later: cached
    acc = wmma4(a, bf, acc);
  }
  for (int v = 0; v < 8; v++) {
    int m = mt * 16 + v + 8 * half;
    keys[(size_t)m * HH + n] = acc[v];
  }
}

// ---------------- Kernel F1: scores[b,l] = v . tanh(q[b]+keys[b,l]) + vb (masked)
__global__ void k_scores(const float* q, const float* keys, const float* vw,
                         const float* vb, const unsigned char* mask,
                         float* scores) {
  int bl = blockIdx.x;
  int b = bl / LL;
  __shared__ float red[256];
  const float* kr = keys + (size_t)bl * HH;
  const float* qr = q + (size_t)b * HH;
  float p = 0.f;
  for (int h = threadIdx.x; h < HH; h += 256)
    p += vw[h] * tanhf(qr[h] + kr[h]);
  red[threadIdx.x] = p; __syncthreads();
  for (int s = 128; s > 0; s >>= 1) {
    if (threadIdx.x < s) red[threadIdx.x] += red[threadIdx.x + s];
    __syncthreads();
  }
  if (threadIdx.x == 0) {
    float sc = red[0] + vb[0];
    if (mask[bl]) sc = -1e20f;
    scores[bl] = sc;
  }
}

// ---------------- Kernel F2: softmax over L, log_softmax, context, build s3
__global__ void k_softctx(const float* scores, const float* enc,
                          const float* hnew, const float* hpar,
                          float* logattn, float* s3) {
  int b = blockIdx.x, tid = threadIdx.x;
  __shared__ float ssc[LL], sattn[LL], red[256];
  __shared__ float sval;
  if (tid < LL) ssc[tid] = scores[b * LL + tid];
  __syncthreads();
  float v = (tid < LL) ? ssc[tid] : -INFINITY;
  red[tid] = v; __syncthreads();
  for (int s = 128; s > 0; s >>= 1) {
    if (tid < s) red[tid] = fmaxf(red[tid], red[tid + s]);
    __syncthreads();
  }
  if (tid == 0) sval = red[0];
  __syncthreads();
  float smax = sval;
  float e = (tid < LL) ? expf(ssc[tid] - smax) : 0.f;
  red[tid] = e; __syncthreads();
  for (int s = 128; s > 0; s >>= 1) {
    if (tid < s) red[tid] += red[tid + s];
    __syncthreads();
  }
  if (tid == 0) sval = red[0];
  __syncthreads();
  float ssum = sval;
  if (tid < LL) {
    sattn[tid] = e / ssum;
    logattn[b * LL + tid] = ssc[tid] - smax - logf(ssum);
  }
  __syncthreads();
  for (int h = tid; h < HH; h += 256) {
    float acc = 0.f;
    const float* eb = enc + (size_t)b * LL * HH + h;
    for (int l = 0; l < LL; l++) acc += sattn[l] * eb[(size_t)l * HH];
    s3[(size_t)b * K3H + h] = acc;                               // context
    s3[(size_t)b * K3H + HH + h] = hnew[(size_t)b * HH + h];     // out
    s3[(size_t)b * K3H + 2 * HH + h] = hpar[(size_t)b * HH + h]; // h_parent
  }
}

// ---------------- Kernel F3: s_t = log_sigmoid([context,out] @ ws_w^T + ws_b)
__global__ void k_switch(const float* s3, const float* wsw, const float* wsb,
                         float* st) {
  int b = blockIdx.x, tid = threadIdx.x;
  __shared__ float red[256];
  const float* row = s3 + (size_t)b * K3H;
  float p = 0.f;
  for (int j = tid; j < 2 * HH; j += 256) p += row[j] * wsw[j];
  red[tid] = p; __syncthreads();
  for (int s = 128; s > 0; s >>= 1) {
    if (tid < s) red[tid] += red[tid + s];
    __syncthreads();
  }
  if (tid == 0) {
    float u = red[0] + wsb[0];
    // stable log-sigmoid
    st[b] = (u >= 0.f) ? -log1pf(expf(-u)) : (u - log1pf(expf(u)));
  }
}

// ---------------- Kernel G: z = s3 @ wg^T + bg -> d_out logits region
// (weight-bandwidth bound: 614 MB of wg_w streamed once, NT hinted)
__global__ void k_vocab(const float* s3, const float* wg, const float* bg,
                        float* out) {
  int wave = (blockIdx.x * blockDim.x + threadIdx.x) >> 5;
  int lane = threadIdx.x & 31;
  const int NT = (VOUT + 15) / 16; // 3126
  if (wave >= NT) return;
  int half = lane >> 4, r = lane & 15;
  int n = wave * 16 + r;
  int nc = (n < VOUT) ? n : (VOUT - 1);   // clamp; junk cols never stored
  float bias = bg[nc];
  v8f acc0, acc1;
  for (int i = 0; i < 8; i++) { acc0[i] = bias; acc1[i] = bias; }
  gemm2_acc(s3, K3H, wg + (size_t)nc * K3H, K3H, half, r, acc0, acc1);
  if (n < VOUT) {
    for (int v = 0; v < 8; v++) {
      int m = v + 8 * half;
      out[(size_t)m * ROWOUT + n] = acc0[v];
      out[(size_t)(16 + m) * ROWOUT + n] = acc1[v];
    }
  }
}

// ---------------- Kernel H: in-place log_softmax + pointer mix over d_out rows
__global__ void k_final(float* out, const float* st, const float* logattn) {
  int b = blockIdx.x, tid = threadIdx.x;
  float* row = out + (size_t)b * ROWOUT;
  __shared__ float red[256];
  __shared__ float sval;
  float m = -INFINITY;
  for (int v = tid; v < VOUT; v += 256) m = fmaxf(m, row[v]);
  red[tid] = m; __syncthreads();
  for (int s = 128; s > 0; s >>= 1) {
    if (tid < s) red[tid] = fmaxf(red[tid], red[tid + s]);
    __syncthreads();
  }
  if (tid == 0) sval = red[0];
  __syncthreads();
  float rowmax = sval;
  float acc = 0.f;
  for (int v = tid; v < VOUT; v += 256) acc += expf(row[v] - rowmax);
  __syncthreads();
  red[tid] = acc; __syncthreads();
  for (int s = 128; s > 0; s >>= 1) {
    if (tid < s) red[tid] += red[tid + s];
    __syncthreads();
  }
  if (tid == 0) sval = rowmax + logf(red[0]);
  __syncthreads();
  float lse = sval;
  float stb = st[b];
  for (int v = tid; v < VOUT; v += 256) row[v] = stb + row[v] - lse;
  float om = -expm1f(stb);
  om = fminf(fmaxf(om, 1e-18f), 1e18f);
  float lc = logf(om);
  for (int l = tid; l < LL; l += 256)
    row[VOUT + l] = lc + logattn[b * LL + l];
}

extern "C" void kernel_launch(void* const* d_in, const int* in_sizes, int n_in,
                              void* d_out, int out_size, void* d_ws,
                              size_t ws_size, hipStream_t stream) {
  (void)in_sizes; (void)n_in; (void)out_size; (void)ws_size;
  const int*   n_input = (const int*)d_in[0];
  const int*   t_input = (const int*)d_in[1];
  const float* h0   = (const float*)d_in[2];
  const float* c0   = (const float*)d_in[3];
  const float* enc  = (const float*)d_in[4];
  const unsigned char* mask = (const unsigned char*)d_in[5];
  const float* hpar = (const float*)d_in[6];
  const float* embN = (const float*)d_in[7];
  const float* embT = (const float*)d_in[8];
  const float* Wih  = (const float*)d_in[9];
  const float* Whh  = (const float*)d_in[10];
  const float* bih  = (const float*)d_in[11];
  const float* bhh  = (const float*)d_in[12];
  const float* Whw  = (const float*)d_in[13];
  const float* Whb  = (const float*)d_in[14];
  const float* Wmw  = (const float*)d_in[15];
  const float* Wmb  = (const float*)d_in[16];
  const float* vw   = (const float*)d_in[17];
  const float* vb   = (const float*)d_in[18];
  const float* wgw  = (const float*)d_in[19];
  const float* wgb  = (const float*)d_in[20];
  const float* wsw  = (const float*)d_in[21];
  const float* wsb  = (const float*)d_in[22];

  float* out  = (float*)d_out;                 // [32, 50131] logits
  float* hout = out + (size_t)BB * ROWOUT;     // [32, 1024] h_new
  float* cout = hout + (size_t)BB * HH;        // [32, 1024] c_new

  // Workspace layout (floats), total ~18 MB
  float* ws      = (float*)d_ws;
  float* x       = ws;                          // 32*768
  float* gates   = x + BB * EIN;                // 32*4096
  float* hnew    = gates + (size_t)BB * G4H;    // 32*1024
  float* q       = hnew + (size_t)BB * HH;      // 32*1024
  float* keys    = q + (size_t)BB * HH;         // 32*128*1024
  float* scores  = keys + (size_t)BB * LL * HH; // 32*128
  float* logattn = scores + BB * LL;            // 32*128
  float* s3      = logattn + BB * LL;           // 32*3072
  float* st      = s3 + (size_t)BB * K3H;       // 32

  k_embed  <<<96,   256, 0, stream>>>(n_input, t_input, embN, embT, x);
  k_gates  <<<32,   256, 0, stream>>>(x, h0, Wih, Whh, bih, bhh, gates); // 256 waves
  k_cell   <<<128,  256, 0, stream>>>(gates, c0, hnew, hout, cout);
  k_query  <<<8,    256, 0, stream>>>(hnew, Whw, Whb, q);                 // 64 waves
  k_keys   <<<2048, 256, 0, stream>>>(enc, Wmw, Wmb, keys);               // 16384 waves
  k_scores <<<BB * LL, 256, 0, stream>>>(q, keys, vw, vb, mask, scores);
  k_softctx<<<BB,   256, 0, stream>>>(scores, enc, hnew, hpar, logattn, s3);
  k_switch <<<BB,   256, 0, stream>>>(s3, wsw, wsb, st);
  k_vocab  <<<391,  256, 0, stream>>>(s3, wgw, wgb, out);                 // 3126 waves
  k_final  <<<BB,   256, 0, stream>>>(out, st, logattn);
}